// NodeLayer_60395830116401
// MI455X (gfx1250) — compile-verified
//
#include <hip/hip_runtime.h>
#include <hip/hip_bf16.h>

#define N_NODES 50000
#define N_EDGES 600000
#define H_DIM   128
#define BN_EPS  1e-5f

typedef __attribute__((ext_vector_type(2))) float v2f;
typedef __attribute__((ext_vector_type(8))) float v8f;

// ---- order-preserving float<->uint encoding for atomic max on signed floats ----
__device__ __forceinline__ unsigned enc_f(float f) {
    unsigned u = __float_as_uint(f);
    return (u & 0x80000000u) ? ~u : (u | 0x80000000u);
}
__device__ __forceinline__ float dec_f(unsigned u) {
    return (u & 0x80000000u) ? __uint_as_float(u ^ 0x80000000u) : __uint_as_float(~u);
}

// ---------------- K0: zero the workspace accumulators ----------------
__global__ void init_ws_kernel(unsigned* __restrict__ smax, float* __restrict__ ssum,
                               float* __restrict__ neigh, float* __restrict__ colsum,
                               float* __restrict__ colsq) {
    size_t i = (size_t)blockIdx.x * blockDim.x + threadIdx.x;
    if (i < N_NODES) { smax[i] = 0u; ssum[i] = 0.0f; }
    if (i < H_DIM)   { colsum[i] = 0.0f; colsq[i] = 0.0f; }
    if (i < (size_t)N_NODES * H_DIM) neigh[i] = 0.0f;
}

// ---------------- K1: per-edge dot product + segment max (1 wave / edge) ----------------
__global__ void edge_score_kernel(const float* __restrict__ emb, const int* __restrict__ src,
                                  const int* __restrict__ dst, float* __restrict__ scores,
                                  unsigned* __restrict__ smax) {
    int e    = (int)((blockIdx.x * blockDim.x + threadIdx.x) >> 5);
    int lane = threadIdx.x & 31;
    if (e >= N_EDGES) return;
    int s = src[e], d = dst[e];
    float4 a = ((const float4*)(emb + (size_t)s * H_DIM))[lane];  // 32 lanes * 4 = 128
    float4 b = ((const float4*)(emb + (size_t)d * H_DIM))[lane];
    float p = a.x * b.x + a.y * b.y + a.z * b.z + a.w * b.w;
    #pragma unroll
    for (int off = 16; off > 0; off >>= 1) p += __shfl_xor(p, off, 32);
    if (lane == 0) {
        scores[e] = p;
        atomicMax(&smax[d], enc_f(p));
    }
}

// ---------------- K2: e = exp(score - segmax[dst]) ; segment sum ----------------
__global__ void edge_exp_kernel(const int* __restrict__ dst, const unsigned* __restrict__ smax,
                                float* __restrict__ scores, float* __restrict__ ssum) {
    int e = blockIdx.x * blockDim.x + threadIdx.x;
    if (e >= N_EDGES) return;
    int d = dst[e];
    unsigned u = smax[d];
    float m = (u == 0u) ? 0.0f : dec_f(u);   // sentinel: node with no in-edges -> 0.0
    float ex = expf(scores[e] - m);
    scores[e] = ex;
    atomicAdd(&ssum[d], ex);
}

// ---------------- K3: scatter msg = e_src * norm into neigh (1 wave / edge) ----------------
__global__ void scatter_kernel(const float* __restrict__ emb, const int* __restrict__ src,
                               const int* __restrict__ dst, const float* __restrict__ ev,
                               const float* __restrict__ ssum, float* __restrict__ neigh) {
    int e    = (int)((blockIdx.x * blockDim.x + threadIdx.x) >> 5);
    int lane = threadIdx.x & 31;
    if (e >= N_EDGES) return;
    int s = src[e], d = dst[e];
    float norm = ev[e] / ssum[d];
    float4 a = ((const float4*)(emb + (size_t)s * H_DIM))[lane];
    float* out = neigh + (size_t)d * H_DIM + lane * 4;
    atomicAdd(out + 0, a.x * norm);
    atomicAdd(out + 1, a.y * norm);
    atomicAdd(out + 2, a.z * norm);
    atomicAdd(out + 3, a.w * norm);
}

// ---------------- K4: h = neigh @ W via V_WMMA_F32_16X16X4_F32 + fused BN stats ----------------
// grid.x = 3125 (M tiles of 16); 8 waves/block, wave w owns N-tile w (16 cols).
// A-frag (16x4 f32): lanes 0-15 rows M=0..15 @ K={k0,k0+1}; lanes 16-31 same rows @ K={k0+2,k0+3}.
// B-frag (4x16 f32): lanes 0-15 cols N @ K={k0,k0+1}; lanes 16-31 cols N @ K={k0+2,k0+3}.
// C/D (16x16 f32): lane<16: vgpr v -> (row v, col lane); lane>=16: (row v+8, col lane-16).
__global__ void __launch_bounds__(256) gemm_bn_kernel(const float* __restrict__ neigh,
                                                      const float* __restrict__ W,
                                                      float* __restrict__ hout,
                                                      float* __restrict__ colsum,
                                                      float* __restrict__ colsq) {
    __shared__ float sW[H_DIM * H_DIM];               // 64 KB of the 320 KB WGP LDS
    for (int i = threadIdx.x; i < H_DIM * H_DIM; i += 256) sW[i] = W[i];
    __syncthreads();

    int wave = threadIdx.x >> 5;
    int lane = threadIdx.x & 31;
    int half = lane >> 4;          // 0: K offset 0, 1: K offset 2
    int l    = lane & 15;
    int row0 = blockIdx.x * 16;    // 50000 = 3125 * 16, exact
    int n0   = wave * 16;

    const float* arow = neigh + (size_t)(row0 + l) * H_DIM;
    v8f c = {};
    #pragma unroll
    for (int k0 = 0; k0 < H_DIM; k0 += 4) {
        int k = k0 + half * 2;
        v2f a, b;
        a.x = arow[k + 0];
        a.y = arow[k + 1];
        b.x = sW[(k + 0) * H_DIM + n0 + l];
        b.y = sW[(k + 1) * H_DIM + n0 + l];
        c = __builtin_amdgcn_wmma_f32_16x16x4_f32(false, a, false, b,
                                                  (short)0, c, false, false);
    }

    // Write h tile and reduce per-column sum / sumsq for BN
    float s = 0.0f, sq = 0.0f;
    int col = n0 + l;
    #pragma unroll
    for (int v = 0; v < 8; ++v) {
        int m = v + half * 8;
        float h = c[v];
        hout[(size_t)(row0 + m) * H_DIM + col] = h;
        s  += h;
        sq += h * h;
    }
    s  += __shfl_xor(s, 16, 32);   // combine rows 0-7 with rows 8-15 of this column
    sq += __shfl_xor(sq, 16, 32);
    if (half == 0) {
        atomicAdd(&colsum[col], s);
        atomicAdd(&colsq[col], sq);
    }
}

// ---------------- K5: finalize BN scale/shift (1 block, 128 threads) ----------------
__global__ void finalize_kernel(const float* __restrict__ colsum, const float* __restrict__ colsq,
                                const float* __restrict__ gamma, const float* __restrict__ beta,
                                float* __restrict__ scale, float* __restrict__ shift) {
    int c = threadIdx.x;
    if (c >= H_DIM) return;
    const float invN = 1.0f / (float)N_NODES;
    float mean = colsum[c] * invN;
    float var  = colsq[c] * invN - mean * mean;   // biased, matches torch BN
    float sc   = gamma[c] * rsqrtf(var + BN_EPS);
    scale[c] = sc;
    shift[c] = beta[c] - mean * sc;
}

// ---------------- K6: out = tanh(h * scale + shift), in place on d_out ----------------
__global__ void apply_kernel(float* __restrict__ out, const float* __restrict__ scale,
                             const float* __restrict__ shift) {
    size_t i = (size_t)blockIdx.x * blockDim.x + threadIdx.x;
    if (i >= (size_t)N_NODES * H_DIM) return;
    int c = (int)(i & (H_DIM - 1));
    out[i] = tanhf(out[i] * scale[c] + shift[c]);
}

extern "C" void kernel_launch(void* const* d_in, const int* in_sizes, int n_in,
                              void* d_out, int out_size, void* d_ws, size_t ws_size,
                              hipStream_t stream) {
    const float* ent_emb = (const float*)d_in[0];   // [N, H]
    const float* neigh_w = (const float*)d_in[1];   // [H, H]
    const float* gamma   = (const float*)d_in[2];   // [H]
    const float* beta    = (const float*)d_in[3];   // [H]
    const int*   src     = (const int*)d_in[4];     // [E]
    const int*   dst     = (const int*)d_in[5];     // [E]
    float*       out     = (float*)d_out;           // [N, H]

    // ---- workspace layout (bytes) ----
    char* ws = (char*)d_ws;
    size_t off = 0;
    float*    scores = (float*)(ws + off);    off += (size_t)N_EDGES * 4;          // edge scores -> e values
    unsigned* smax   = (unsigned*)(ws + off); off += (size_t)N_NODES * 4;          // encoded segment max
    float*    ssum   = (float*)(ws + off);    off += (size_t)N_NODES * 4;          // segment sum
    float*    neigh  = (float*)(ws + off);    off += (size_t)N_NODES * H_DIM * 4;  // aggregated messages
    float*    colsum = (float*)(ws + off);    off += (size_t)H_DIM * 4;
    float*    colsq  = (float*)(ws + off);    off += (size_t)H_DIM * 4;
    float*    scale  = (float*)(ws + off);    off += (size_t)H_DIM * 4;
    float*    shift  = (float*)(ws + off);    off += (size_t)H_DIM * 4;
    (void)ws_size; (void)in_sizes; (void)n_in; (void)out_size;

    const size_t NH = (size_t)N_NODES * H_DIM;

    // K0: zero accumulators (must re-zero every call; no cross-call state)
    {
        int threads = 256;
        int blocks  = (int)((NH + threads - 1) / threads);
        init_ws_kernel<<<blocks, threads, 0, stream>>>(smax, ssum, neigh, colsum, colsq);
    }
    // K1: edge dot products + segment max (1 wave32 per edge)
    {
        int threads = 256;                                   // 8 waves
        int blocks  = (N_EDGES * 32 + threads - 1) / threads;
        edge_score_kernel<<<blocks, threads, 0, stream>>>(ent_emb, src, dst, scores, smax);
    }
    // K2: exp + segment sum (1 thread per edge)
    {
        int threads = 256;
        int blocks  = (N_EDGES + threads - 1) / threads;
        edge_exp_kernel<<<blocks, threads, 0, stream>>>(dst, smax, scores, ssum);
    }
    // K3: normalized message scatter-add (1 wave32 per edge)
    {
        int threads = 256;
        int blocks  = (N_EDGES * 32 + threads - 1) / threads;
        scatter_kernel<<<blocks, threads, 0, stream>>>(ent_emb, src, dst, scores, ssum, neigh);
    }
    // K4: WMMA f32 GEMM (h = neigh @ W) writing h into d_out + fused column stats
    {
        gemm_bn_kernel<<<N_NODES / 16, 256, 0, stream>>>(neigh, neigh_w, out, colsum, colsq);
    }
    // K5: BN scale/shift
    {
        finalize_kernel<<<1, 128, 0, stream>>>(colsum, colsq, gamma, beta, scale, shift);
    }
    // K6: normalize + tanh, in place
    {
        int threads = 256;
        int blocks  = (int)((NH + threads - 1) / threads);
        apply_kernel<<<blocks, threads, 0, stream>>>(out, scale, shift);
    }
}